// S2DTW_67388036874600
// MI455X (gfx1250) — compile-verified
//
#include <hip/hip_runtime.h>

// Soft-DTW contrastive loss (b=128, n=32, d=512), fp32 throughout.
// GEMM per batch-pair via V_WMMA_F32_16X16X4_F32 (exact fp32 path),
// DTW wavefront via wave32 shuffles + LDS-staged cost tile.

typedef float v2f __attribute__((ext_vector_type(2)));
typedef float v8f __attribute__((ext_vector_type(8)));

#define SGAMMA    0.1f
#define INV_GAMMA 10.0f
#define FBIG      1e10f

constexpr int BB = 128;   // batch
constexpr int NN = 32;    // sequence length
constexpr int DD = 512;   // embedding dim

// One wave per (video i, text j) pair: 32x32x512 GEMM -> cost tile -> soft-DTW.
__global__ __launch_bounds__(32)
void sdtw_pair_kernel(const float* __restrict__ video,
                      const float* __restrict__ text,
                      float* __restrict__ neg /* [BB*BB] = -sdtw */) {
  __shared__ float lds[NN * 33];             // 33-float pitch: conflict-free diagonals

  const int j    = blockIdx.x;               // text batch index
  const int i    = blockIdx.y;               // video batch index
  const int lane = threadIdx.x;
  const int lh   = lane >> 4;                // lane half selects K-pair (K+0/1 vs K+2/3)
  const int ln   = lane & 15;                // M (for A) / N (for B) within a 16-tile

  // A fragment rows: video rows; B fragment cols: text rows (C = V * T^T).
  const float* pA0 = video + ((size_t)i * NN + ln) * DD + 2 * lh;
  const float* pA1 = pA0 + 16 * DD;
  const float* pB0 = text  + ((size_t)j * NN + ln) * DD + 2 * lh;
  const float* pB1 = pB0 + 16 * DD;

  v8f c00 = {}, c01 = {}, c10 = {}, c11 = {};
#pragma unroll 4
  for (int k = 0; k < DD; k += 4) {
    v2f a0 = *(const v2f*)(pA0 + k);
    v2f a1 = *(const v2f*)(pA1 + k);
    v2f b0 = *(const v2f*)(pB0 + k);
    v2f b1 = *(const v2f*)(pB1 + k);
    c00 = __builtin_amdgcn_wmma_f32_16x16x4_f32(false, a0, false, b0, (short)0, c00, false, false);
    c01 = __builtin_amdgcn_wmma_f32_16x16x4_f32(false, a0, false, b1, (short)0, c01, false, false);
    c10 = __builtin_amdgcn_wmma_f32_16x16x4_f32(false, a1, false, b0, (short)0, c10, false, false);
    c11 = __builtin_amdgcn_wmma_f32_16x16x4_f32(false, a1, false, b1, (short)0, c11, false, false);
  }

  // C/D layout: VGPR r -> M = r (lanes 0-15) or r+8 (lanes 16-31); N = lane&15.
  // Write cost = -dot into LDS.
#pragma unroll
  for (int r = 0; r < 8; ++r) {
    const int m0 = r + 8 * lh;
    lds[m0 * 33 + ln]             = -c00[r];
    lds[m0 * 33 + 16 + ln]        = -c01[r];
    lds[(m0 + 16) * 33 + ln]      = -c10[r];
    lds[(m0 + 16) * 33 + 16 + ln] = -c11[r];
  }
  __syncthreads();

  // Anti-diagonal soft-DTW scan. Lane l owns R row i_r = l+1.
  // dm2/dm1 hold diagonals k-2 and k-1 (value R[i_r, k - i_r]).
  float dm2 = FBIG, dm1 = FBIG;             // diag k=0 and k=1 for rows >= 1
#pragma unroll 1
  for (int k = 2; k <= 2 * NN; ++k) {
    float up2 = __shfl_up(dm2, 1, 32);      // R[i_r-1, k-i_r-1]
    float up1 = __shfl_up(dm1, 1, 32);      // R[i_r-1, k-i_r]
    if (lane == 0) {                        // row 0 boundary: R[0,0]=0, else BIG
      up2 = (k == 2) ? 0.0f : FBIG;
      up1 = FBIG;
    }
    const int  jr     = k - (lane + 1);     // column of R for this lane
    const bool active = (jr >= 1) && (jr <= NN);
    const float c = active ? lds[lane * 33 + (jr - 1)] : FBIG;
    // softmin_gamma(up2, up1, dm1), numerically stable
    float m = fminf(up2, fminf(up1, dm1));
    float s = __expf((m - up2) * INV_GAMMA) +
              __expf((m - up1) * INV_GAMMA) +
              __expf((m - dm1) * INV_GAMMA);
    float dk = c + (m - SGAMMA * __logf(s));
    dk  = active ? dk : FBIG;
    dm2 = dm1;
    dm1 = dk;
  }
  if (lane == 31)                            // lane 31 -> i_r = 32, R[32,32]
    neg[i * BB + j] = -dm1;
}

// Per-row logsumexp_j(neg[r, :]) - neg[r, r]; one wave per row.
__global__ __launch_bounds__(32)
void row_lse_kernel(const float* __restrict__ neg, float* __restrict__ rowv) {
  const int r = blockIdx.x;
  const int l = threadIdx.x;
  float x0 = neg[r * BB + l];
  float x1 = neg[r * BB + l + 32];
  float x2 = neg[r * BB + l + 64];
  float x3 = neg[r * BB + l + 96];
  float m = fmaxf(fmaxf(x0, x1), fmaxf(x2, x3));
#pragma unroll
  for (int d = 16; d >= 1; d >>= 1) m = fmaxf(m, __shfl_xor(m, d, 32));
  float s = __expf(x0 - m) + __expf(x1 - m) + __expf(x2 - m) + __expf(x3 - m);
#pragma unroll
  for (int d = 16; d >= 1; d >>= 1) s += __shfl_xor(s, d, 32);
  if (l == 0) rowv[r] = m + __logf(s) - neg[r * BB + r];
}

// Mean of 128 row values -> scalar output.
__global__ __launch_bounds__(32)
void mean_kernel(const float* __restrict__ rowv, float* __restrict__ out) {
  const int l = threadIdx.x;
  float v = rowv[l] + rowv[l + 32] + rowv[l + 64] + rowv[l + 96];
#pragma unroll
  for (int d = 16; d >= 1; d >>= 1) v += __shfl_xor(v, d, 32);
  if (l == 0) out[0] = v * (1.0f / (float)BB);
}

extern "C" void kernel_launch(void* const* d_in, const int* in_sizes, int n_in,
                              void* d_out, int out_size, void* d_ws, size_t ws_size,
                              hipStream_t stream) {
  const float* video = (const float*)d_in[0];   // [128, 32, 512] fp32
  const float* text  = (const float*)d_in[1];   // [128, 32, 512] fp32
  float* wsNeg = (float*)d_ws;                  // [128*128]
  float* wsRow = wsNeg + BB * BB;               // [128]
  float* out   = (float*)d_out;                 // scalar fp32

  dim3 grid1(BB, BB);
  sdtw_pair_kernel<<<grid1, 32, 0, stream>>>(video, text, wsNeg);
  row_lse_kernel<<<BB, 32, 0, stream>>>(wsNeg, wsRow);
  mean_kernel<<<1, 32, 0, stream>>>(wsRow, out);
}